// LinearDispatch_9354438770889
// MI455X (gfx1250) — compile-verified
//
#include <hip/hip_runtime.h>

// Problem constants (match reference)
#define B_ROWS 8192
#define D_IN   512
#define D_OUT  512
#define NCLS   16

#define ROWS_PER_TILE 32
#define KCHUNK   256
#define A_STRIDE 260     // 256 + 4 pad: 16B-aligned rows, conflict-free b64 frag loads
#define MAX_TILES (B_ROWS / ROWS_PER_TILE + NCLS)   // 272

typedef __attribute__((ext_vector_type(2))) float v2f;
typedef __attribute__((ext_vector_type(4))) float v4f;
typedef __attribute__((ext_vector_type(8))) float v8f;

// Workspace layout (int units)
#define WS_COUNTS   0            // 16
#define WS_OFFSETS  16           // 17 (row offset per class, +total)
#define WS_CURSORS  33           // 16 (scatter cursors)
#define WS_TILEBASE 49           // 17 (cumulative M-tile base per class, +total)
#define WS_ROWLIST  66           // B_ROWS row indices grouped by class

__global__ void ld_zero_meta(int* __restrict__ ws) {
    int t = threadIdx.x;
    if (t < NCLS) ws[WS_COUNTS + t] = 0;
}

__global__ void ld_count(const int* __restrict__ cls, int* __restrict__ ws) {
    int i = blockIdx.x * 256 + threadIdx.x;
    if (i < B_ROWS) atomicAdd(&ws[WS_COUNTS + cls[i]], 1);
}

__global__ void ld_scan(int* __restrict__ ws) {
    if (threadIdx.x == 0) {
        int off = 0, tb = 0;
        for (int c = 0; c < NCLS; ++c) {
            ws[WS_OFFSETS  + c] = off;
            ws[WS_CURSORS  + c] = off;
            ws[WS_TILEBASE + c] = tb;
            int cnt = ws[WS_COUNTS + c];
            off += cnt;
            tb  += (cnt + ROWS_PER_TILE - 1) / ROWS_PER_TILE;
        }
        ws[WS_OFFSETS  + NCLS] = off;
        ws[WS_TILEBASE + NCLS] = tb;
    }
}

__global__ void ld_scatter(const int* __restrict__ cls, int* __restrict__ ws) {
    int i = blockIdx.x * 256 + threadIdx.x;
    if (i < B_ROWS) {
        int c = cls[i];
        int p = atomicAdd(&ws[WS_CURSORS + c], 1);
        ws[WS_ROWLIST + p] = i;
    }
}

// Grouped GEMM: one block = one 32-row M-group of one class, all 512 output cols.
// 8 waves (wave32); wave w owns cols [w*64, w*64+64) = 4 WMMA N-tiles,
// and BOTH 16-row M-tiles of the group -> every B fragment feeds 2 WMMAs.
// K is staged in two 256-wide LDS phases; accumulators persist.
__global__ __launch_bounds__(256)
void ld_gemm(const float* __restrict__ x, const float* __restrict__ W,
             const float* __restrict__ bias, const int* __restrict__ ws,
             float* __restrict__ out) {
    __shared__ float Alds[ROWS_PER_TILE * A_STRIDE];   // 33,280 B

    const int t = blockIdx.x;
    const int total = ws[WS_TILEBASE + NCLS];
    if (t >= total) return;

    // Find owning class: largest c with t >= tileBase[c]
    int c = 0;
#pragma unroll
    for (int cc = 0; cc < NCLS; ++cc) {
        if (t >= ws[WS_TILEBASE + cc + 1]) c = cc + 1;
    }
    const int mrel    = t - ws[WS_TILEBASE + c];
    const int rowBase = ws[WS_OFFSETS + c] + mrel * ROWS_PER_TILE;
    const int rowEnd  = ws[WS_OFFSETS + c + 1];

    const int tid  = threadIdx.x;
    const int wave = tid >> 5;
    const int lane = tid & 31;
    const int m16  = lane & 15;            // M for A frag / N for B,C,D frags
    const int koff = (lane >> 4) * 2;      // lanes 16-31 hold K+2,K+3
    const int n0   = wave * 64;

    const float* __restrict__ Wc = W + (size_t)c * (D_IN * D_OUT);
    const float* __restrict__ bc = bias + c * D_OUT;

    // Bias depends only on N (C/D layout: VGPR v -> M=v+8*(lane>>4), N=m16).
    float bb0 = bc[n0 +  0 + m16];
    float bb1 = bc[n0 + 16 + m16];
    float bb2 = bc[n0 + 32 + m16];
    float bb3 = bc[n0 + 48 + m16];
    v8f a00, a01, a02, a03;   // M-tile 0 (rows 0-15)  x N-tiles 0-3
    v8f a10, a11, a12, a13;   // M-tile 1 (rows 16-31) x N-tiles 0-3
#pragma unroll
    for (int v = 0; v < 8; ++v) {
        a00[v] = bb0; a01[v] = bb1; a02[v] = bb2; a03[v] = bb3;
        a10[v] = bb0; a11[v] = bb1; a12[v] = bb2; a13[v] = bb3;
    }

    for (int kc = 0; kc < D_IN; kc += KCHUNK) {
        __syncthreads();   // previous phase fully consumed before restaging
        // Stage 32 rows x 256 K into LDS; 2048 float4 chunks / 256 threads = 8 each.
#pragma unroll
        for (int it = 0; it < 8; ++it) {
            int idx = it * 256 + tid;      // 0..2047
            int r   = idx >> 6;            // 64 float4 per row
            int c4  = idx & 63;
            v4f val = {0.f, 0.f, 0.f, 0.f};
            int gr  = rowBase + r;
            if (gr < rowEnd) {
                int grow = ws[WS_ROWLIST + gr];
                val = *(const v4f*)(x + (size_t)grow * D_IN + kc + c4 * 4);
            }
            *(v4f*)(&Alds[r * A_STRIDE + c4 * 4]) = val;   // 16B aligned
        }
        __syncthreads();

        const float* ap0 = &Alds[m16 * A_STRIDE + koff];
        const float* ap1 = ap0 + 16 * A_STRIDE;
        int widx = (kc + koff) * D_OUT + n0 + m16;   // += 4*512 per k-step

#pragma unroll 2
        for (int k0 = 0; k0 < KCHUNK; k0 += 4) {
            v2f fa0 = *(const v2f*)ap0;    // A frag, M-tile 0 (16x4 f32 layout)
            v2f fa1 = *(const v2f*)ap1;    // A frag, M-tile 1
            ap0 += 4; ap1 += 4;

            v2f b0, b1, b2, b3;            // B frags (4x16 f32 layout), coalesced b32
            b0.x = Wc[widx +  0];   b0.y = Wc[widx + D_OUT +  0];
            b1.x = Wc[widx + 16];   b1.y = Wc[widx + D_OUT + 16];
            b2.x = Wc[widx + 32];   b2.y = Wc[widx + D_OUT + 32];
            b3.x = Wc[widx + 48];   b3.y = Wc[widx + D_OUT + 48];
            widx += 4 * D_OUT;

            a00 = __builtin_amdgcn_wmma_f32_16x16x4_f32(false, fa0, false, b0, (short)0, a00, false, false);
            a10 = __builtin_amdgcn_wmma_f32_16x16x4_f32(false, fa1, false, b0, (short)0, a10, false, false);
            a01 = __builtin_amdgcn_wmma_f32_16x16x4_f32(false, fa0, false, b1, (short)0, a01, false, false);
            a11 = __builtin_amdgcn_wmma_f32_16x16x4_f32(false, fa1, false, b1, (short)0, a11, false, false);
            a02 = __builtin_amdgcn_wmma_f32_16x16x4_f32(false, fa0, false, b2, (short)0, a02, false, false);
            a12 = __builtin_amdgcn_wmma_f32_16x16x4_f32(false, fa1, false, b2, (short)0, a12, false, false);
            a03 = __builtin_amdgcn_wmma_f32_16x16x4_f32(false, fa0, false, b3, (short)0, a03, false, false);
            a13 = __builtin_amdgcn_wmma_f32_16x16x4_f32(false, fa1, false, b3, (short)0, a13, false, false);
        }
    }

    // Store D: VGPR v -> M = v + 8*(lane>>4), N = n0 + tileN*16 + m16.
    const int mhalf = (lane >> 4) * 8;
#pragma unroll
    for (int v = 0; v < 8; ++v) {
        int m  = v + mhalf;
        int gr = rowBase + m;             // M-tile 0
        if (gr < rowEnd) {
            int grow = ws[WS_ROWLIST + gr];
            float* __restrict__ orow = out + (size_t)grow * D_OUT + n0 + m16;
            orow[ 0] = a00[v];
            orow[16] = a01[v];
            orow[32] = a02[v];
            orow[48] = a03[v];
        }
        int gr1 = rowBase + 16 + m;       // M-tile 1
        if (gr1 < rowEnd) {
            int grow = ws[WS_ROWLIST + gr1];
            float* __restrict__ orow = out + (size_t)grow * D_OUT + n0 + m16;
            orow[ 0] = a10[v];
            orow[16] = a11[v];
            orow[32] = a12[v];
            orow[48] = a13[v];
        }
    }
}

extern "C" void kernel_launch(void* const* d_in, const int* in_sizes, int n_in,
                              void* d_out, int out_size, void* d_ws, size_t ws_size,
                              hipStream_t stream) {
    const float* x    = (const float*)d_in[0];   // [B, D_IN] f32
    const int*   cls  = (const int*)  d_in[1];   // [B] i32
    const float* W    = (const float*)d_in[2];   // [C, D_IN, D_OUT] f32
    const float* bias = (const float*)d_in[3];   // [C, D_OUT] f32
    float*       out  = (float*)d_out;           // [B, D_OUT] f32
    int*         ws   = (int*)d_ws;

    (void)in_sizes; (void)n_in; (void)out_size; (void)ws_size;

    ld_zero_meta<<<1, 64, 0, stream>>>(ws);
    ld_count   <<<B_ROWS / 256, 256, 0, stream>>>(cls, ws);
    ld_scan    <<<1, 32, 0, stream>>>(ws);
    ld_scatter <<<B_ROWS / 256, 256, 0, stream>>>(cls, ws);
    ld_gemm    <<<MAX_TILES, 256, 0, stream>>>(x, W, bias, ws, out);
}